// EnhancedBitcoinGCN_75557064671729
// MI455X (gfx1250) — compile-verified
//
#include <hip/hip_runtime.h>
#include <cstdint>
#include <cstddef>

// ---------------------------------------------------------------------------
// EnhancedBitcoinGCN on MI455X (gfx1250).
// fp32 end-to-end; dense GEMMs via V_WMMA_F32_16X16X4_F32 (wave32 WMMA).
// Pipeline is memory/scatter bound (~1.5-2 GB regular traffic + ~130M f32
// atomics), so fp32 WMMA is already far above the bandwidth roofline.
//
// Round-3 fix vs round-2: cap accumulator tiles per wave at NT<=4 by
// splitting NOUT=128 across two column-waves (32 acc VGPRs/wave instead of
// 64). The software pipeliner no longer spills accumulators to scratch or
// emits rotation movs in the K-loop.
// ---------------------------------------------------------------------------

typedef __attribute__((ext_vector_type(2))) float v2f;
typedef __attribute__((ext_vector_type(8))) float v8f;

#define IN_F 164

// ------------------------- degree / dinv -----------------------------------
__global__ void deg_init_kernel(float* deg, int n) {
  int i = blockIdx.x * blockDim.x + threadIdx.x;
  if (i < n) deg[i] = 1.0f;  // self-loop
}

__global__ void deg_accum_kernel(const int* __restrict__ dst, float* deg, int e) {
  int i = blockIdx.x * blockDim.x + threadIdx.x;
  if (i < e) atomicAdd(&deg[dst[i]], 1.0f);
}

__global__ void deg_rsqrt_kernel(float* deg, int n) {
  int i = blockIdx.x * blockDim.x + threadIdx.x;
  if (i < n) deg[i] = rsqrtf(deg[i]);
}

// ------------------------- WMMA fp32 GEMM ----------------------------------
// Block = 256 threads = 8 waves = (8/WN) row-groups x WN column-waves.
// Each wave computes 16 rows x (NT*16) cols, K stepped by 4 with
// v_wmma_f32_16x16x4_f32. NT <= 4 keeps accumulators at <=32 VGPRs/wave.
//
// Fragment layouts per CDNA5 ISA 7.12.2 (32-bit, wave32):
//   A 16x4 : lane L holds A[L&15][2*(L>>4)] , A[L&15][2*(L>>4)+1]
//   B 4x16 : lane L holds B[2*(L>>4)][L&15] , B[2*(L>>4)+1][L&15]
//   C/D    : lane L, vgpr j holds D[j + 8*(L>>4)][L&15]
//
// LDS holds W as k-pairs: lds_wp[p*NOUT + n] = { W[2p][n], W[2p+1][n] }.
//
// EPI 0: out0[r,n] = relu(acc + bias[n])
// EPI 1: out0[r,n] = acc (xw) ; out1[r,n] = acc * dinv[r]^2 (self-loop agg init)
template<int K, int NOUT, int EPI>
__global__ void wmma_gemm_kernel(const float* __restrict__ X,
                                 const float* __restrict__ W,
                                 const float* __restrict__ bias,
                                 const float* __restrict__ dinv,
                                 float* __restrict__ out0,
                                 float* __restrict__ out1,
                                 int nrows) {
  static_assert(K % 4 == 0 && NOUT % 16 == 0, "tile shape");
  constexpr int WN = (NOUT >= 128) ? 2 : 1;  // column-waves per row-group
  constexpr int NT = NOUT / (16 * WN);       // accumulator tiles per wave (<=4)
  constexpr int RG = 8 / WN;                 // row-groups per block

  extern __shared__ float smem[];            // K*NOUT floats, viewed as v2f pairs
  v2f* lds_wp = (v2f*)smem;
  const int tid = threadIdx.x;
  for (int i = tid; i < (K / 2) * NOUT; i += 256) {
    const int kp = i / NOUT;                 // NOUT is a power of two -> shift
    const int n = i - kp * NOUT;
    v2f w;
    w.x = W[(2 * kp) * NOUT + n];
    w.y = W[(2 * kp + 1) * NOUT + n];
    lds_wp[i] = w;
  }
  __syncthreads();

  const int wave = tid >> 5;
  const int lane = tid & 31;
  const int rowGroup = wave >> (WN - 1);     // WN==1: wave ; WN==2: wave/2
  const int colHalf = wave & (WN - 1);
  const int rowBase = (blockIdx.x * RG + rowGroup) * 16;
  const int colBase = colHalf * (NT * 16);
  if (rowBase >= nrows) return;  // wave-uniform exit; EXEC stays all-ones for WMMA

  const int mA = lane & 15;       // A-matrix row / B,C,D column within tile
  const int khalf = lane >> 4;    // which k-pair of the 4-deep step
  const int rowA = rowBase + mA;
  // Clamp instead of predicating: garbage A rows only affect D rows >= nrows,
  // which are never stored. Keeps loads unconditional (no exec divergence).
  const int rowLd = rowA < nrows ? rowA : (nrows - 1);
  const float* xrow = X + (size_t)rowLd * K + khalf * 2;  // 8B-aligned (K even)
  const v2f* bbase = lds_wp + (size_t)khalf * NOUT + colBase + mA;

  v8f acc[NT];
#pragma unroll
  for (int t = 0; t < NT; ++t)
#pragma unroll
    for (int j = 0; j < 8; ++j) acc[t][j] = 0.0f;

  for (int kk = 0; kk < K; kk += 4) {
    const v2f a = *(const v2f*)(xrow + kk);            // one global_load_b64
    const v2f* bp = bbase + (size_t)(kk >> 1) * NOUT;  // pair row kk/2 + khalf
#pragma unroll
    for (int t = 0; t < NT; ++t) {
      const v2f b = bp[t * 16];                        // one ds_load_b64
      acc[t] = __builtin_amdgcn_wmma_f32_16x16x4_f32(
          /*neg_a=*/false, a, /*neg_b=*/false, b,
          /*c_mod=*/(short)0, acc[t], /*reuse_a=*/false, /*reuse_b=*/false);
    }
  }

#pragma unroll
  for (int t = 0; t < NT; ++t) {
    const int n = colBase + t * 16 + mA;
#pragma unroll
    for (int j = 0; j < 8; ++j) {
      const int r = rowBase + j + 8 * khalf;  // C/D row mapping
      if (r < nrows) {
        const float val = acc[t][j];
        if (EPI == 0) {
          out0[(size_t)r * NOUT + n] = fmaxf(val + bias[n], 0.0f);
        } else {
          out0[(size_t)r * NOUT + n] = val;
          const float s = dinv[r];
          out1[(size_t)r * NOUT + n] = val * s * s;
        }
      }
    }
  }
}

// ------------------------- LayerNorm (wave per row) ------------------------
// If prebias != nullptr: v = relu(v + prebias[c]) before normalization.
template<int C>
__global__ void layernorm_kernel(const float* __restrict__ in,
                                 const float* __restrict__ prebias,
                                 const float* __restrict__ g,
                                 const float* __restrict__ b,
                                 float* __restrict__ out, int n) {
  const int gid = blockIdx.x * blockDim.x + threadIdx.x;
  const int row = gid >> 5;
  const int lane = gid & 31;
  if (row >= n) return;
  constexpr int PER = C / 32;
  const float* p = in + (size_t)row * C;
  float vals[PER];
  float s = 0.0f;
#pragma unroll
  for (int i = 0; i < PER; ++i) {
    const int c = lane + 32 * i;
    float v = p[c];
    if (prebias) v = fmaxf(v + prebias[c], 0.0f);
    vals[i] = v;
    s += v;
  }
#pragma unroll
  for (int off = 16; off > 0; off >>= 1) s += __shfl_xor(s, off, 32);
  const float mean = s * (1.0f / C);
  float vs = 0.0f;
#pragma unroll
  for (int i = 0; i < PER; ++i) { const float d = vals[i] - mean; vs += d * d; }
#pragma unroll
  for (int off = 16; off > 0; off >>= 1) vs += __shfl_xor(vs, off, 32);
  const float inv = rsqrtf(vs * (1.0f / C) + 1e-5f);
  float* q = out + (size_t)row * C;
#pragma unroll
  for (int i = 0; i < PER; ++i) {
    const int c = lane + 32 * i;
    q[c] = (vals[i] - mean) * inv * g[c] + b[c];
  }
}

// ------------------------- edge scatter (wave per edge) --------------------
template<int C>
__global__ void scatter_kernel(const int* __restrict__ src, const int* __restrict__ dst,
                               const float* __restrict__ xw, const float* __restrict__ dinv,
                               float* __restrict__ agg, int e) {
  const int gid = blockIdx.x * blockDim.x + threadIdx.x;
  const int edge = gid >> 5;
  const int lane = gid & 31;
  if (edge >= e) return;
  const int s = src[edge], d = dst[edge];
  const float w = dinv[s] * dinv[d];
  const float* xs = xw + (size_t)s * C;
  float* ad = agg + (size_t)d * C;
#pragma unroll
  for (int c = lane; c < C; c += 32) atomicAdd(&ad[c], xs[c] * w);
}

// ------------------------- residual: h1 = relu(agg + b) + h ----------------
__global__ void residual_relu_kernel(const float* __restrict__ agg,
                                     const float* __restrict__ bias,
                                     const float* __restrict__ h,
                                     float* __restrict__ out, int total) {
  const int i = blockIdx.x * blockDim.x + threadIdx.x;
  if (i < total) {
    const int c = i & 127;  // C == 128
    out[i] = fmaxf(agg[i] + bias[c], 0.0f) + h[i];
  }
}

// ------------------------- fused head (thread per node) --------------------
// h3 = relu(agg3 + bg3);  mha(seq=1) == (h3@Wv+bv)@Wo+bo (softmax of 1 key
// is exactly 1, q/k cancel);  then relu(@Wc1), relu(@Wc2), @Wc3.
__global__ void head_kernel(const float* __restrict__ agg3,
                            const float* __restrict__ bg3,
                            const float* __restrict__ Wv, const float* __restrict__ bv,
                            const float* __restrict__ Wo, const float* __restrict__ bo,
                            const float* __restrict__ Wc1, const float* __restrict__ bc1,
                            const float* __restrict__ Wc2, const float* __restrict__ bc2,
                            const float* __restrict__ Wc3, const float* __restrict__ bc3,
                            float* __restrict__ out, int n) {
  __shared__ float sWv[32 * 32], sWo[32 * 32], sWc1[32 * 16], sWc2[16 * 32];
  __shared__ float sWc3[32], sbg3[32], sbv[32], sbo[32], sbc1[16], sbc2[32];
  __shared__ float sbc3;
  const int tid = threadIdx.x;
  for (int i = tid; i < 1024; i += 256) { sWv[i] = Wv[i]; sWo[i] = Wo[i]; }
  for (int i = tid; i < 512; i += 256) { sWc1[i] = Wc1[i]; sWc2[i] = Wc2[i]; }
  if (tid < 32) {
    sWc3[tid] = Wc3[tid]; sbg3[tid] = bg3[tid]; sbv[tid] = bv[tid];
    sbo[tid] = bo[tid];   sbc2[tid] = bc2[tid];
  }
  if (tid < 16) sbc1[tid] = bc1[tid];
  if (tid == 0) sbc3 = bc3[0];
  __syncthreads();

  const int node = blockIdx.x * blockDim.x + tid;
  if (node >= n) return;

  float a[32], v[32];
#pragma unroll
  for (int i = 0; i < 32; ++i) a[i] = fmaxf(agg3[(size_t)node * 32 + i] + sbg3[i], 0.0f);
#pragma unroll
  for (int j = 0; j < 32; ++j) {
    float s = sbv[j];
#pragma unroll
    for (int i = 0; i < 32; ++i) s += a[i] * sWv[i * 32 + j];
    v[j] = s;
  }
#pragma unroll
  for (int j = 0; j < 32; ++j) {  // o -> a
    float s = sbo[j];
#pragma unroll
    for (int i = 0; i < 32; ++i) s += v[i] * sWo[i * 32 + j];
    a[j] = s;
  }
  float c1[16];
#pragma unroll
  for (int j = 0; j < 16; ++j) {
    float s = sbc1[j];
#pragma unroll
    for (int i = 0; i < 32; ++i) s += a[i] * sWc1[i * 16 + j];
    c1[j] = fmaxf(s, 0.0f);
  }
#pragma unroll
  for (int j = 0; j < 32; ++j) {  // c2 -> v
    float s = sbc2[j];
#pragma unroll
    for (int i = 0; i < 16; ++i) s += c1[i] * sWc2[i * 32 + j];
    v[j] = fmaxf(s, 0.0f);
  }
  float s = sbc3;
#pragma unroll
  for (int i = 0; i < 32; ++i) s += v[i] * sWc3[i];
  out[node] = s;
}

// ---------------------------------------------------------------------------
extern "C" void kernel_launch(void* const* d_in, const int* in_sizes, int n_in,
                              void* d_out, int out_size, void* d_ws, size_t ws_size,
                              hipStream_t stream) {
  const float* x   = (const float*)d_in[0];
  const int*   ei  = (const int*)d_in[1];
  const float* Wi  = (const float*)d_in[2];
  const float* bi  = (const float*)d_in[3];
  const float* g1  = (const float*)d_in[4];
  const float* b1  = (const float*)d_in[5];
  const float* Wg1 = (const float*)d_in[6];
  const float* bg1 = (const float*)d_in[7];
  const float* Wg2 = (const float*)d_in[8];
  const float* bg2 = (const float*)d_in[9];
  const float* g2  = (const float*)d_in[10];
  const float* b2  = (const float*)d_in[11];
  const float* Wg3 = (const float*)d_in[12];
  const float* bg3 = (const float*)d_in[13];
  // d_in[14..17] = Wq,bq,Wk,bk -- mathematically unused (softmax over 1 key == 1)
  const float* Wv  = (const float*)d_in[18];
  const float* bv  = (const float*)d_in[19];
  const float* Wo  = (const float*)d_in[20];
  const float* bo  = (const float*)d_in[21];
  const float* Wc1 = (const float*)d_in[22];
  const float* bc1 = (const float*)d_in[23];
  const float* Wc2 = (const float*)d_in[24];
  const float* bc2 = (const float*)d_in[25];
  const float* Wc3 = (const float*)d_in[26];
  const float* bc3 = (const float*)d_in[27];
  float* out = (float*)d_out;

  const int N = in_sizes[0] / IN_F;
  const int E = in_sizes[1] / 2;
  const int* srcIdx = ei;
  const int* dstIdx = ei + E;

  // workspace layout: dinv[N] | B0[N*128] | B1[N*128] | B2[N*128]
  float* ws = (float*)d_ws;
  float* dinv = ws;
  const size_t off = ((size_t)N + 255) & ~(size_t)255;
  float* B0 = ws + off;
  float* B1 = B0 + (size_t)N * 128;
  float* B2 = B1 + (size_t)N * 128;

  const dim3 blk(256);
  const int gemmBlocks64rows = (N + 63) / 64;    // NOUT=128 kernels (2 col-waves)
  const int gemmBlocks128rows = (N + 127) / 128; // NOUT<=64 kernels
  const int rowWaveBlocks = ((N * 32) + 255) / 256;
  const int edgeWaveBlocks = ((E * 32) + 255) / 256;

  // degrees -> dinv
  deg_init_kernel<<<(N + 255) / 256, blk, 0, stream>>>(dinv, N);
  deg_accum_kernel<<<(E + 255) / 256, blk, 0, stream>>>(dstIdx, dinv, E);
  deg_rsqrt_kernel<<<(N + 255) / 256, blk, 0, stream>>>(dinv, N);

  // h_pre = relu(x @ Wi + bi)  ->  B0
  wmma_gemm_kernel<IN_F, 128, 0><<<gemmBlocks64rows, blk, IN_F * 128 * sizeof(float), stream>>>(
      x, Wi, bi, nullptr, B0, nullptr, N);
  // h = LN(h_pre)  ->  B1
  layernorm_kernel<128><<<rowWaveBlocks, blk, 0, stream>>>(B0, nullptr, g1, b1, B1, N);

  // GCN layer 1: xw -> B0, agg(self-loop) -> B2, scatter, h1 = relu(agg+b)+h -> B0
  wmma_gemm_kernel<128, 128, 1><<<gemmBlocks64rows, blk, 128 * 128 * sizeof(float), stream>>>(
      B1, Wg1, nullptr, dinv, B0, B2, N);
  scatter_kernel<128><<<edgeWaveBlocks, blk, 0, stream>>>(srcIdx, dstIdx, B0, dinv, B2, E);
  residual_relu_kernel<<<((N * 128) + 255) / 256, blk, 0, stream>>>(B2, bg1, B1, B0, N * 128);

  // GCN layer 2: xw2 -> B1, agg2 -> B2, scatter, h2 = LN(relu(agg2+bg2)) -> B0
  wmma_gemm_kernel<128, 64, 1><<<gemmBlocks128rows, blk, 128 * 64 * sizeof(float), stream>>>(
      B0, Wg2, nullptr, dinv, B1, B2, N);
  scatter_kernel<64><<<edgeWaveBlocks, blk, 0, stream>>>(srcIdx, dstIdx, B1, dinv, B2, E);
  layernorm_kernel<64><<<rowWaveBlocks, blk, 0, stream>>>(B2, bg2, g2, b2, B0, N);

  // GCN layer 3: xw3 -> B1, agg3 -> B2, scatter
  wmma_gemm_kernel<64, 32, 1><<<gemmBlocks128rows, blk, 64 * 32 * sizeof(float), stream>>>(
      B0, Wg3, nullptr, dinv, B1, B2, N);
  scatter_kernel<32><<<edgeWaveBlocks, blk, 0, stream>>>(srcIdx, dstIdx, B1, dinv, B2, E);

  // fused head: relu(agg3+bg3) -> mha(seq1) -> MLP -> out
  head_kernel<<<(N + 255) / 256, blk, 0, stream>>>(
      B2, bg3, Wv, bv, Wo, bo, Wc1, bc1, Wc2, bc2, Wc3, bc3, out, N);

  (void)n_in; (void)out_size; (void)ws_size;
}